// LaCore_32427003085189
// MI455X (gfx1250) — compile-verified
//
#include <hip/hip_runtime.h>
#include <hip/hip_bf16.h>

typedef __attribute__((ext_vector_type(2))) float v2f;
typedef __attribute__((ext_vector_type(8))) float v8f;

#define NGRAPH 64
#define NCLUST 25000
#define HDIM   128
#define NCLS   10

// ---------------------------------------------------------------------------
// Preprocessing: per-graph node counts, per-cluster node counts, pooled batch
// ---------------------------------------------------------------------------
__global__ void k_node_prep(const int* __restrict__ batch, const int* __restrict__ cluster,
                            float* __restrict__ cntB, float* __restrict__ cntC,
                            int* __restrict__ batch_p, int n) {
  int i = blockIdx.x * blockDim.x + threadIdx.x;
  if (i >= n) return;
  int b = batch[i];
  int c = cluster[i];
  atomicAdd(&cntB[b], 1.0f);
  atomicAdd(&cntC[c], 1.0f);
  atomicMax(&batch_p[c], b);           // batch ids >= 0; init 0 == maximum(.,0)
}

// Pooled edge endpoints + degree counts (node-level and cluster-level)
__global__ void k_edge_prep(const int* __restrict__ ei, const int* __restrict__ cluster,
                            int* __restrict__ srcP, int* __restrict__ dstP,
                            float* __restrict__ degN, float* __restrict__ degC, int e_) {
  int e = blockIdx.x * blockDim.x + threadIdx.x;
  if (e >= e_) return;
  int s = ei[e];
  int d = ei[e_ + e];
  int sp = cluster[s], dp = cluster[d];
  srcP[e] = sp; dstP[e] = dp;
  atomicAdd(&degN[d], 1.0f);
  atomicAdd(&degC[dp], 1.0f);
}

__global__ void k_finalize_dis(float* __restrict__ dis, int n) {
  int i = blockIdx.x * blockDim.x + threadIdx.x;
  if (i < n) dis[i] = rsqrtf(dis[i] + 1.0f);   // deg^-1/2 with self loop
}

// ---------------------------------------------------------------------------
// GEMM: out[M,128] = A[M,128] @ W[128,128] via V_WMMA_F32_16X16X4_F32
// one block = 16 rows; 8 waves, each owning a 16-wide column tile
// ---------------------------------------------------------------------------
__global__ __launch_bounds__(256)
void k_gemm128_wmma(const float* __restrict__ A, const float* __restrict__ W,
                    float* __restrict__ out, int M) {
  __shared__ float ldsA[16 * 128];
  const int tid  = threadIdx.x;
  const int wave = tid >> 5;
  const int lane = tid & 31;
  const int tileM = blockIdx.x * 16;

  // stage 16x128 A strip in LDS (clamp tail rows; stores are guarded later)
  for (int i = tid; i < 16 * 128; i += 256) {
    int r = i >> 7, c = i & 127;
    int row = tileM + r;
    if (row >= M) row = M - 1;
    ldsA[i] = A[(size_t)row * 128 + c];
  }
  __syncthreads();

  const int col    = wave * 16 + (lane & 15);
  const int hiHalf = lane >> 4;          // 0: K{0,1}, 1: K{2,3}
  const int r      = lane & 15;          // A-matrix row (M) for this lane

  v8f acc = {};
  for (int k = 0; k < 128; k += 4) {
    const int ka = k + hiHalf * 2;
    v2f a, b;
    a.x = ldsA[r * 128 + ka];
    a.y = ldsA[r * 128 + ka + 1];
    b.x = W[(size_t)ka * 128 + col];
    b.y = W[(size_t)(ka + 1) * 128 + col];
    acc = __builtin_amdgcn_wmma_f32_16x16x4_f32(false, a, false, b,
                                                (short)0, acc, false, false);
  }

  // C/D layout: VGPR v, lanes 0-15 -> M=v, lanes 16-31 -> M=v+8, N=lane&15
#pragma unroll
  for (int v = 0; v < 8; ++v) {
    int row = tileM + v + hiHalf * 8;
    if (row < M) out[(size_t)row * 128 + col] = acc[v];
  }
}

// ---------------------------------------------------------------------------
// GCN pieces
// ---------------------------------------------------------------------------
// out = h * dis^2 + bias   (self-loop term; must run before the edge scatter)
__global__ void k_selfloop(const float* __restrict__ h, const float* __restrict__ dis,
                           const float* __restrict__ bias, float* __restrict__ out, int total) {
  int i = blockIdx.x * blockDim.x + threadIdx.x;
  if (i >= total) return;
  int row = i >> 7, f = i & 127;
  float dd = dis[row];
  out[i] = h[i] * dd * dd + bias[f];
}

// out[dst] += h[src] * dis[src]*dis[dst]   — one wave per edge, float4 per lane
__global__ void k_scatter(const float* __restrict__ h, const int* __restrict__ srcI,
                          const int* __restrict__ dstI, const float* __restrict__ dis,
                          float* __restrict__ out, int e_) {
  int t = blockIdx.x * blockDim.x + threadIdx.x;
  int e = t >> 5, lane = t & 31;
  if (e >= e_) return;
  int s = srcI[e], d = dstI[e];
  float w = dis[s] * dis[d];
  float4 v = ((const float4*)(h + (size_t)s * 128))[lane];
  float* o = out + (size_t)d * 128 + lane * 4;
  atomicAdd(o + 0, v.x * w);
  atomicAdd(o + 1, v.y * w);
  atomicAdd(o + 2, v.z * w);
  atomicAdd(o + 3, v.w * w);
}

__global__ void k_relu(float* __restrict__ x, int n) {
  int i = blockIdx.x * blockDim.x + threadIdx.x;
  if (i < n) x[i] = fmaxf(x[i], 0.0f);
}

// sum + max pooling by segment id (values are post-ReLU >= 0 -> int-bit max OK)
__global__ void k_pool_graph(const float* __restrict__ h, const int* __restrict__ seg,
                             float* __restrict__ sum, int* __restrict__ maxI, int n) {
  int t = blockIdx.x * blockDim.x + threadIdx.x;
  int node = t >> 5, lane = t & 31;
  if (node >= n) return;
  int g = seg[node];
  float4 v = ((const float4*)(h + (size_t)node * 128))[lane];
  float* s = sum + (size_t)g * 128 + lane * 4;
  int*  mx = maxI + (size_t)g * 128 + lane * 4;
  atomicAdd(s + 0, v.x); atomicMax(mx + 0, __float_as_int(v.x));
  atomicAdd(s + 1, v.y); atomicMax(mx + 1, __float_as_int(v.y));
  atomicAdd(s + 2, v.z); atomicMax(mx + 2, __float_as_int(v.z));
  atomicAdd(s + 3, v.w); atomicMax(mx + 3, __float_as_int(v.w));
}

// sum-only pooling (cluster mean numerator)
__global__ void k_pool_sum(const float* __restrict__ h, const int* __restrict__ seg,
                           float* __restrict__ sum, int n) {
  int t = blockIdx.x * blockDim.x + threadIdx.x;
  int node = t >> 5, lane = t & 31;
  if (node >= n) return;
  int c = seg[node];
  float4 v = ((const float4*)(h + (size_t)node * 128))[lane];
  float* s = sum + (size_t)c * 128 + lane * 4;
  atomicAdd(s + 0, v.x);
  atomicAdd(s + 1, v.y);
  atomicAdd(s + 2, v.z);
  atomicAdd(s + 3, v.w);
}

__global__ void k_divrows(float* __restrict__ x, const float* __restrict__ cnt, int total) {
  int i = blockIdx.x * blockDim.x + threadIdx.x;
  if (i >= total) return;
  x[i] /= fmaxf(cnt[i >> 7], 1.0f);
}

__global__ void k_count_bp(const int* __restrict__ batch_p, float* __restrict__ cntBP, int c) {
  int i = blockIdx.x * blockDim.x + threadIdx.x;
  if (i < c) atomicAdd(&cntBP[batch_p[i]], 1.0f);
}

// ---------------------------------------------------------------------------
// Head
// ---------------------------------------------------------------------------
__global__ void k_build_z(const float* __restrict__ preSum, const int* __restrict__ preMaxI,
                          const float* __restrict__ postSum, const int* __restrict__ postMaxI,
                          const float* __restrict__ cntB, const float* __restrict__ cntBP,
                          float* __restrict__ zbuf) {
  int idx = blockIdx.x * blockDim.x + threadIdx.x;
  if (idx >= NGRAPH * 512) return;
  int g = idx >> 9, i = idx & 511;
  int part = i >> 7, f = i & 127;
  float v;
  if      (part == 0) v = preSum[g * 128 + f] / fmaxf(cntB[g], 1.0f);
  else if (part == 1) v = __int_as_float(preMaxI[g * 128 + f]);
  else if (part == 2) v = postSum[g * 128 + f] / fmaxf(cntBP[g], 1.0f);
  else                v = __int_as_float(postMaxI[g * 128 + f]);
  zbuf[idx] = v;
}

__global__ void k_head1(const float* __restrict__ zbuf, const float* __restrict__ w,
                        const float* __restrict__ b, float* __restrict__ z1) {
  int g = blockIdx.x;      // 64
  int j = threadIdx.x;     // 128
  float acc = b[j];
  for (int i = 0; i < 512; ++i) acc += zbuf[g * 512 + i] * w[i * 128 + j];
  z1[g * 128 + j] = fmaxf(acc, 0.0f);
}

__global__ void k_head2(const float* __restrict__ z1, const float* __restrict__ w,
                        const float* __restrict__ b, float* __restrict__ out) {
  int g = blockIdx.x * blockDim.x + threadIdx.x;
  if (g >= NGRAPH) return;
  float v[NCLS];
  for (int k = 0; k < NCLS; ++k) v[k] = b[k];
  for (int i = 0; i < 128; ++i) {
    float zi = z1[g * 128 + i];
    for (int k = 0; k < NCLS; ++k) v[k] += zi * w[i * NCLS + k];
  }
  float m = v[0];
  for (int k = 1; k < NCLS; ++k) m = fmaxf(m, v[k]);
  float s = 0.0f;
  for (int k = 0; k < NCLS; ++k) s += expf(v[k] - m);
  float ls = logf(s);
  for (int k = 0; k < NCLS; ++k) out[g * NCLS + k] = v[k] - m - ls;
}

// ---------------------------------------------------------------------------
extern "C" void kernel_launch(void* const* d_in, const int* in_sizes, int n_in,
                              void* d_out, int out_size, void* d_ws, size_t ws_size,
                              hipStream_t stream) {
  const float* x       = (const float*)d_in[0];
  const int*   ei      = (const int*)d_in[1];     // [2,E]: ei[e]=src, ei[E+e]=dst
  const int*   batch   = (const int*)d_in[2];
  const int*   cluster = (const int*)d_in[3];
  const float* W1  = (const float*)d_in[6];  const float* b1  = (const float*)d_in[7];
  const float* W2  = (const float*)d_in[8];  const float* b2  = (const float*)d_in[9];
  const float* W3  = (const float*)d_in[10]; const float* b3  = (const float*)d_in[11];
  const float* l1w = (const float*)d_in[12]; const float* l1b = (const float*)d_in[13];
  const float* l2w = (const float*)d_in[14]; const float* l2b = (const float*)d_in[15];

  const int N = in_sizes[0] / HDIM;
  const int E = in_sizes[1] / 2;
  const int C = NCLUST;

  // ---- workspace carve-up (256B aligned) ----
  char* wp = (char*)d_ws;
  auto carve = [&](size_t bytes) -> char* {
    char* p = wp;
    wp += (bytes + 255) & ~(size_t)255;
    return p;
  };
  int*   srcP     = (int*)  carve((size_t)E * 4);
  int*   dstP     = (int*)  carve((size_t)E * 4);
  int*   batch_p  = (int*)  carve((size_t)C * 4);
  float* disN     = (float*)carve((size_t)N * 4);
  float* disC     = (float*)carve((size_t)C * 4);
  float* cntB     = (float*)carve(NGRAPH * 4);
  float* cntC     = (float*)carve((size_t)C * 4);
  float* cntBP    = (float*)carve(NGRAPH * 4);
  float* preSum   = (float*)carve(NGRAPH * 128 * 4);
  int*   preMaxI  = (int*)  carve(NGRAPH * 128 * 4);
  float* postSum  = (float*)carve(NGRAPH * 128 * 4);
  int*   postMaxI = (int*)  carve(NGRAPH * 128 * 4);
  float* zbuf     = (float*)carve(NGRAPH * 512 * 4);
  float* z1       = (float*)carve(NGRAPH * 128 * 4);
  float* bufH     = (float*)carve((size_t)N * 128 * 4);  // GEMM output (xW / hpW)
  float* bufA     = (float*)carve((size_t)N * 128 * 4);  // conv1 out -> h1
  float* hpA      = (float*)carve((size_t)C * 128 * 4);  // hp0 / hp2
  float* hpB      = (float*)carve((size_t)C * 128 * 4);  // hp1

  // ---- zero accumulators (every call; graph-capture safe) ----
  hipMemsetAsync(disN, 0, (size_t)N * 4, stream);
  hipMemsetAsync(disC, 0, (size_t)C * 4, stream);
  hipMemsetAsync(cntB, 0, NGRAPH * 4, stream);
  hipMemsetAsync(cntC, 0, (size_t)C * 4, stream);
  hipMemsetAsync(cntBP, 0, NGRAPH * 4, stream);
  hipMemsetAsync(batch_p, 0, (size_t)C * 4, stream);
  hipMemsetAsync(preSum, 0, NGRAPH * 128 * 4, stream);
  hipMemsetAsync(preMaxI, 0, NGRAPH * 128 * 4, stream);
  hipMemsetAsync(postSum, 0, NGRAPH * 128 * 4, stream);
  hipMemsetAsync(postMaxI, 0, NGRAPH * 128 * 4, stream);
  hipMemsetAsync(hpA, 0, (size_t)C * 128 * 4, stream);

  const int T = 256;
  auto nb = [](long long n, int t) { return (unsigned)((n + t - 1) / t); };

  // preprocessing
  k_node_prep<<<nb(N, T), T, 0, stream>>>(batch, cluster, cntB, cntC, batch_p, N);
  k_edge_prep<<<nb(E, T), T, 0, stream>>>(ei, cluster, srcP, dstP, disN, disC, E);
  k_finalize_dis<<<nb(N, T), T, 0, stream>>>(disN, N);
  k_finalize_dis<<<nb(C, T), T, 0, stream>>>(disC, C);

  // conv1: h = x@W1 ; out = scatter + selfloop + b1 ; relu
  k_gemm128_wmma<<<nb(N, 16), T, 0, stream>>>(x, W1, bufH, N);
  k_selfloop<<<nb((long long)N * 128, T), T, 0, stream>>>(bufH, disN, b1, bufA, N * 128);
  k_scatter<<<nb((long long)E * 32, T), T, 0, stream>>>(bufH, ei, ei + E, disN, bufA, E);
  k_relu<<<nb((long long)N * 128, T), T, 0, stream>>>(bufA, N * 128);

  // pre pools + cluster mean pool
  k_pool_graph<<<nb((long long)N * 32, T), T, 0, stream>>>(bufA, batch, preSum, preMaxI, N);
  k_pool_sum<<<nb((long long)N * 32, T), T, 0, stream>>>(bufA, cluster, hpA, N);
  k_divrows<<<nb((long long)C * 128, T), T, 0, stream>>>(hpA, cntC, C * 128);

  // conv2 on pooled graph
  k_gemm128_wmma<<<nb(C, 16), T, 0, stream>>>(hpA, W2, bufH, C);
  k_selfloop<<<nb((long long)C * 128, T), T, 0, stream>>>(bufH, disC, b2, hpB, C * 128);
  k_scatter<<<nb((long long)E * 32, T), T, 0, stream>>>(bufH, srcP, dstP, disC, hpB, E);
  k_relu<<<nb((long long)C * 128, T), T, 0, stream>>>(hpB, C * 128);

  // conv3 on pooled graph
  k_gemm128_wmma<<<nb(C, 16), T, 0, stream>>>(hpB, W3, bufH, C);
  k_selfloop<<<nb((long long)C * 128, T), T, 0, stream>>>(bufH, disC, b3, hpA, C * 128);
  k_scatter<<<nb((long long)E * 32, T), T, 0, stream>>>(bufH, srcP, dstP, disC, hpA, E);
  k_relu<<<nb((long long)C * 128, T), T, 0, stream>>>(hpA, C * 128);

  // post pools
  k_count_bp<<<nb(C, T), T, 0, stream>>>(batch_p, cntBP, C);
  k_pool_graph<<<nb((long long)C * 32, T), T, 0, stream>>>(hpA, batch_p, postSum, postMaxI, C);

  // head
  k_build_z<<<nb(NGRAPH * 512, T), T, 0, stream>>>(preSum, preMaxI, postSum, postMaxI,
                                                   cntB, cntBP, zbuf);
  k_head1<<<NGRAPH, 128, 0, stream>>>(zbuf, l1w, l1b, z1);
  k_head2<<<1, NGRAPH, 0, stream>>>(z1, l2w, l2b, (float*)d_out);
}